// VectorQuantizer_12206297055307
// MI455X (gfx1250) — compile-verified
//
#include <hip/hip_runtime.h>

typedef __attribute__((ext_vector_type(2))) float v2f;
typedef __attribute__((ext_vector_type(8))) float v8f;
typedef __attribute__((ext_vector_type(4))) unsigned int v4u;
typedef __attribute__((ext_vector_type(8))) int v8i;
typedef __attribute__((ext_vector_type(4))) int v4i;

#define KCODES   512
#define DIM      64
#define BB       32
#define TTLEN    4096
#define NTOT     (BB * TTLEN)        // 131072 vectors
#define ZQ_ELEMS (BB * DIM * TTLEN)  // 8388608
#define TILES    (NTOT / 16)         // 8192
#define WPB      16                  // waves per block (512 threads)
#define CBS_STRIDE 66                // padded LDS row stride (dwords): 64 + 2 pad

// ---------------------------------------------------------------------------
// Kernel 1: per-code squared norms -> ws[0..511]
// ---------------------------------------------------------------------------
__global__ void vq_cnorm_kernel(const float* __restrict__ cb,
                                float* __restrict__ ws) {
    int t = threadIdx.x;  // 512 threads, 1 block
    const float* row = cb + (size_t)t * DIM;
    float s = 0.f;
#pragma unroll
    for (int j = 0; j < DIM; ++j) s += row[j] * row[j];
    ws[t] = s;
}

// ---------------------------------------------------------------------------
// TDM: stage the whole codebook (512 x 64 f32) into LDS with a single
// TENSOR_LOAD_TO_LDS.  D# per CDNA5 ISA:
//   group0: count=1, lds_addr, global_addr, type=2
//   group1: data_size=4B, pad_enable=1, pad_interval=64DW(code 5),
//           pad_amount=2DW(code 1)  -> LDS row stride 66 dwords
//           tensor_dim0=64, tensor_dim1=512, tile_dim0=64, tile_dim1=512,
//           tensor_dim0_stride=64
// ---------------------------------------------------------------------------
__device__ inline void tdm_load_codebook(const float* cb, float* cbs) {
    unsigned lds = (unsigned)(unsigned long long)cbs;   // low 32 bits = LDS byte offset
    unsigned long long ga = (unsigned long long)cb;
    v4u g0 = { 1u,                                   // count=1, user descriptor
               lds,                                  // lds_addr
               (unsigned)ga,                         // global_addr[31:0]
               (unsigned)((ga >> 32) & 0x1FFFFFFu) | (2u << 30) };  // addr[56:32] | type=2
    v8i g1 = { (int)((2u << 16)        // data_size = 4 bytes
                   | (1u << 20)        // pad_enable
                   | (5u << 22)        // pad_interval: 64 dwords
                   | (1u << 25)),      // pad_amount: 2 dwords
               (int)(64u << 16),       // tensor_dim0 = 64 (low16 in [31:16])
               (int)(512u << 16),      // tensor_dim1 = 512
               (int)(64u << 16),       // tile_dim0 = 64
               (int)512,               // tile_dim1 = 512, tile_dim2 = 0
               (int)64,                // tensor_dim0_stride = 64
               0, 0 };
    v4i g2 = {0, 0, 0, 0};
    v4i g3 = {0, 0, 0, 0};
    v8i g4 = {0, 0, 0, 0, 0, 0, 0, 0};
    __builtin_amdgcn_tensor_load_to_lds(g0, g1, g2, g3, g4, 0);
}

// ---------------------------------------------------------------------------
// Kernel 2: main VQ kernel. One wave32 handles 16 consecutive t of one batch b.
// dist(row, code) via chained V_WMMA_F32_16X16X4_F32:
//   A row = [-2*z (64 dims), 1]   B col = [c (64 dims), ||c||^2]
// => acc = ||c||^2 - 2 z.c  (||z||^2 is constant per row -> argmin unchanged)
// ---------------------------------------------------------------------------
__global__ void __launch_bounds__(32 * WPB)
vq_main_kernel(const float* __restrict__ z,
               const float* __restrict__ cb,
               float* __restrict__ ws,
               float* __restrict__ out) {
    __shared__ float cbs[KCODES * CBS_STRIDE];  // padded codebook, 132 KB
    __shared__ float lz[WPB * DIM * 16];        // per-wave staged z tile [d*16+tt]
    __shared__ int   lidx[WPB * 16];            // per-wave chosen code per row

    const int wave = threadIdx.x >> 5;
    const int lane = threadIdx.x & 31;
    const int tile = blockIdx.x * WPB + wave;   // 0..8191
    const int b    = tile >> 8;                 // 256 tiles per batch
    const int t0   = (tile & 255) << 4;

    // Wave 0 kicks off the async codebook DMA; it overlaps the z staging below.
    if (wave == 0) tdm_load_codebook(cb, cbs);

    const float* zbase = z + (size_t)b * DIM * TTLEN + t0;
    float* lzw = lz + wave * DIM * 16;

    // Stage z[b, :, t0:t0+16] -> LDS (coalesced: 16 contiguous floats per d)
#pragma unroll
    for (int i = 0; i < 32; ++i) {
        int e = i * 32 + lane;
        int d = e >> 4, tt = e & 15;
        lzw[e] = zbase[(size_t)d * TTLEN + tt];
    }
    if (wave == 0) __builtin_amdgcn_s_wait_tensorcnt(0);
    __syncthreads();

    const int M  = lane & 15;   // A: row index / B,D: column index
    const int hi = lane >> 4;   // K sub-pair select

    // A fragments, 17 K-steps (K = 0..67). 32-bit A 16x4 layout:
    // lanes 0-15 hold K={4k,4k+1}, lanes 16-31 hold K={4k+2,4k+3}.
    v2f afrag[17];
#pragma unroll
    for (int k = 0; k < 16; ++k) {
        int d0 = 4 * k + 2 * hi;
        v2f a;
        a.x = -2.0f * lzw[d0 * 16 + M];
        a.y = -2.0f * lzw[(d0 + 1) * 16 + M];
        afrag[k] = a;
    }
    {   // K=64 -> 1.0 (only lanes hi==0), K=65..67 -> 0
        v2f a; a.x = hi ? 0.f : 1.f; a.y = 0.f; afrag[16] = a;
    }

    float bestd[8];
    int   besti[8];
#pragma unroll
    for (int r = 0; r < 8; ++r) { bestd[r] = 3.4e38f; besti[r] = 0; }

    for (int n0 = 0; n0 < KCODES; n0 += 16) {
        // B fragments from padded LDS: stride 66 => 16 lanes hit distinct banks,
        // even dword offset keeps ds_load_b64 8-byte aligned.
        const float* brow = cbs + (n0 + M) * CBS_STRIDE + 2 * hi;

        v8f acc = {0.f, 0.f, 0.f, 0.f, 0.f, 0.f, 0.f, 0.f};
#pragma unroll
        for (int k = 0; k < 16; ++k) {
            v2f bv = *(const v2f*)(brow + 4 * k);
            acc = __builtin_amdgcn_wmma_f32_16x16x4_f32(
                false, afrag[k], false, bv, (short)0, acc, false, false);
        }
        {   // norm step: B col gets ||c||^2 at K=64
            v2f bl; bl.x = hi ? 0.f : ws[n0 + M]; bl.y = 0.f;
            acc = __builtin_amdgcn_wmma_f32_16x16x4_f32(
                false, afrag[16], false, bl, (short)0, acc, false, false);
        }
        // D layout: VGPR r, lanes 0-15: row=r, col=lane; lanes 16-31: row=r+8
        int cidx = n0 + M;
#pragma unroll
        for (int r = 0; r < 8; ++r) {
            float dcur = acc[r];
            if (dcur < bestd[r]) { bestd[r] = dcur; besti[r] = cidx; }
        }
    }

    // Cross-lane argmin across the 16 N-lanes of each half (rows r+8*hi).
    // Tie-break: lower code index (matches jnp.argmin).
#pragma unroll
    for (int r = 0; r < 8; ++r) {
        float dv = bestd[r]; int iv = besti[r];
#pragma unroll
        for (int m = 1; m <= 8; m <<= 1) {
            float od = __shfl_xor(dv, m, 16);
            int   oi = __shfl_xor(iv, m, 16);
            if (od < dv || (od == dv && oi < iv)) { dv = od; iv = oi; }
        }
        bestd[r] = dv; besti[r] = iv;
    }
    if (M == 0) {  // lanes 0 and 16 hold final results for rows 0-7 / 8-15
#pragma unroll
        for (int r = 0; r < 8; ++r) {
            int row = r + 8 * hi;
            lidx[wave * 16 + row] = besti[r];
            out[ZQ_ELEMS + (size_t)b * TTLEN + t0 + row] = (float)besti[r];
        }
    }
    __syncthreads();

    // Gather z_q from LDS codebook, scatter to (B, D, T), accumulate sq error.
    float lsum = 0.f;
#pragma unroll
    for (int i = 0; i < 32; ++i) {
        int e = i * 32 + lane;
        int d = e >> 4, tt = e & 15;
        int ci = lidx[wave * 16 + tt];
        float q  = cbs[ci * CBS_STRIDE + d];
        float df = lzw[e] - q;
        lsum += df * df;
        out[(size_t)b * DIM * TTLEN + (size_t)d * TTLEN + t0 + tt] = q;
    }
#pragma unroll
    for (int m = 1; m < 32; m <<= 1) lsum += __shfl_xor(lsum, m, 32);
    if (lane == 0) ws[KCODES + tile] = lsum;
}

// ---------------------------------------------------------------------------
// Kernel 3: deterministic fixed-order loss reduction.
// vq_loss = 1.25 * mean((z - z_q)^2)   (codebook + 0.25*commitment, identical)
// ---------------------------------------------------------------------------
__global__ void vq_loss_reduce_kernel(const float* __restrict__ ws,
                                      float* __restrict__ out) {
    __shared__ float sp[256];
    int t = threadIdx.x;
    float s = 0.f;
    for (int i = t; i < TILES; i += 256) s += ws[KCODES + i];
    sp[t] = s;
    __syncthreads();
    for (int off = 128; off > 0; off >>= 1) {
        if (t < off) sp[t] += sp[t + off];
        __syncthreads();
    }
    if (t == 0)
        out[ZQ_ELEMS + NTOT] = 1.25f * sp[0] / (float)ZQ_ELEMS;
}

// ---------------------------------------------------------------------------
extern "C" void kernel_launch(void* const* d_in, const int* in_sizes, int n_in,
                              void* d_out, int out_size, void* d_ws, size_t ws_size,
                              hipStream_t stream) {
    const float* z  = (const float*)d_in[0];   // (32, 64, 4096) f32
    const float* cb = (const float*)d_in[1];   // (512, 64) f32
    float* out = (float*)d_out;                // [z_q | indices | loss]
    float* ws  = (float*)d_ws;                 // [cnorm(512) | partials(8192)]

    vq_cnorm_kernel<<<1, KCODES, 0, stream>>>(cb, ws);
    vq_main_kernel<<<TILES / WPB, 32 * WPB, 0, stream>>>(z, cb, ws, out);
    vq_loss_reduce_kernel<<<1, 256, 0, stream>>>(ws, out);
}